// ImprovedGCNBlock_57449482551791
// MI455X (gfx1250) — compile-verified
//
#include <hip/hip_runtime.h>
#include <math.h>

#define CCH 128
#define LDSS 132  // padded LDS row stride (floats) -> conflict-free float2 reads

typedef float v2f __attribute__((ext_vector_type(2)));
typedef float v8f __attribute__((ext_vector_type(8)));

__device__ __forceinline__ float lrelu(float e) { return e > 0.f ? e : 0.2f * e; }

// mixed-sign float atomic max via integer atomics (emits global_atomic_max_i32 / min_u32)
__device__ __forceinline__ void atomic_max_f32(float* addr, float val) {
  if (val >= 0.f)
    atomicMax((int*)addr, __float_as_int(val));
  else
    atomicMin((unsigned int*)addr, __float_as_uint(val));
}

// ---------------------------------------------------------------------------
// K1: xw = x @ W^T  via V_WMMA_F32_16X16X4_F32.  M=N nodes, N=128, K=128.
// Block = 256 threads (8 waves). Block covers 128 rows; each wave a 16-row band.
// W (64KB) staged in LDS with padded stride.
// A frag (lane l): rows 16, lane m=l&15, hi=l>>4 holds x[row][4k+2hi .. +1]  (float2)
// B frag (lane l): holds W[j0 + (l&15)][4k + 2hi .. +1]                      (float2)
// D frag: VGPR v -> row base+v+8*hi, col j0+(l&15)
// ---------------------------------------------------------------------------
__global__ void k_gemm(const float* __restrict__ x, const float* __restrict__ W,
                       float* __restrict__ xw, int Nn) {
  __shared__ float lw[CCH * LDSS];
  const int tid = threadIdx.x;
  for (int i = tid; i < CCH * CCH; i += 256) {
    int r = i >> 7, c = i & 127;
    lw[r * LDSS + c] = W[i];
  }
  __syncthreads();

  const int wave = tid >> 5;
  const int lane = tid & 31;
  const int hi   = lane >> 4;
  const int mrow = lane & 15;
  const int row_base = blockIdx.x * 128 + wave * 16;
  if (row_base >= Nn) return;  // N % 16 == 0 -> tiles are all-or-nothing

  const float* arow = x + (size_t)(row_base + mrow) * CCH + 2 * hi;
  v2f a[32];
#pragma unroll
  for (int kk = 0; kk < 32; ++kk)
    a[kk] = *(const v2f*)(arow + kk * 4);

#pragma unroll
  for (int jt = 0; jt < 8; ++jt) {
    const int j0 = jt * 16;
    v8f c = {0.f, 0.f, 0.f, 0.f, 0.f, 0.f, 0.f, 0.f};
    const float* brow = lw + (j0 + mrow) * LDSS + 2 * hi;
#pragma unroll
    for (int kk = 0; kk < 32; ++kk) {
      v2f b = *(const v2f*)(brow + kk * 4);
      c = __builtin_amdgcn_wmma_f32_16x16x4_f32(false, a[kk], false, b,
                                                (short)0, c, false, false);
    }
    float* orow = xw + (size_t)(row_base + 8 * hi) * CCH + j0 + mrow;
#pragma unroll
    for (int v = 0; v < 8; ++v)
      orow[(size_t)v * CCH] = c[v];
  }
}

// ---------------------------------------------------------------------------
// K2: per-node (one wave/node): a_s, a_d; m[n] = leakyrelu(a_s+a_d) (self-loop
// seed, so m is always finite); denom[n] = 0; agg row zeroed.
// ---------------------------------------------------------------------------
__global__ void k_prep(const float* __restrict__ xw, const float* __restrict__ att_src,
                       const float* __restrict__ att_dst, float* __restrict__ as_,
                       float* __restrict__ ad_, float* __restrict__ mx,
                       float* __restrict__ dn, float* __restrict__ agg, int Nn) {
  const int wid  = (int)((blockIdx.x * (size_t)blockDim.x + threadIdx.x) >> 5);
  const int lane = threadIdx.x & 31;
  if (wid >= Nn) return;

  const float4 v  = *(const float4*)(xw + (size_t)wid * CCH + lane * 4);
  const float4 ws = *(const float4*)(att_src + lane * 4);
  const float4 wd = *(const float4*)(att_dst + lane * 4);
  float ps = v.x * ws.x + v.y * ws.y + v.z * ws.z + v.w * ws.w;
  float pd = v.x * wd.x + v.y * wd.y + v.z * wd.z + v.w * wd.w;
#pragma unroll
  for (int off = 16; off > 0; off >>= 1) {
    ps += __shfl_xor(ps, off);
    pd += __shfl_xor(pd, off);
  }
  const float4 z = {0.f, 0.f, 0.f, 0.f};
  *(float4*)(agg + (size_t)wid * CCH + lane * 4) = z;
  if (lane == 0) {
    as_[wid] = ps;
    ad_[wid] = pd;
    mx[wid]  = lrelu(ps + pd);  // self-loop edge score
    dn[wid]  = 0.f;
  }
}

// ---------------------------------------------------------------------------
// K3: per-edge segment max (thread per edge)
// ---------------------------------------------------------------------------
__global__ void k_emax(const long long* __restrict__ src, const long long* __restrict__ dst,
                       const float* __restrict__ as_, const float* __restrict__ ad_,
                       float* __restrict__ mx, int En) {
  const int i = blockIdx.x * blockDim.x + threadIdx.x;
  if (i >= En) return;
  const int s = (int)src[i], d = (int)dst[i];
  atomic_max_f32(mx + d, lrelu(as_[s] + ad_[d]));
}

// ---------------------------------------------------------------------------
// K4: per-edge weighted scatter (one wave/edge): w = exp(e - m[d]);
// denom[d] += w (lane 0); agg[d] += w * xw[s]  (coalesced 512B gather/scatter)
// ---------------------------------------------------------------------------
__global__ void k_escatter(const long long* __restrict__ src, const long long* __restrict__ dst,
                           const float* __restrict__ as_, const float* __restrict__ ad_,
                           const float* __restrict__ mx, float* __restrict__ dn,
                           const float* __restrict__ xw, float* __restrict__ agg, int En) {
  const int wid  = (int)((blockIdx.x * (size_t)blockDim.x + threadIdx.x) >> 5);
  const int lane = threadIdx.x & 31;
  if (wid >= En) return;
  const int s = (int)src[wid], d = (int)dst[wid];
  const float w = expf(lrelu(as_[s] + ad_[d]) - mx[d]);
  if (lane == 0) unsafeAtomicAdd(dn + d, w);
  const float4 v = *(const float4*)(xw + (size_t)s * CCH + lane * 4);
  float* ag = agg + (size_t)d * CCH + lane * 4;
  unsafeAtomicAdd(ag + 0, w * v.x);
  unsafeAtomicAdd(ag + 1, w * v.y);
  unsafeAtomicAdd(ag + 2, w * v.z);
  unsafeAtomicAdd(ag + 3, w * v.w);
}

// ---------------------------------------------------------------------------
// K5: finalize (one wave/node): add self-loop term, normalize, +bias,
// GroupNorm1 -> exact GELU -> +x residual -> GroupNorm2 -> out.
// Lane l holds channels 4l..4l+3; group g (16 ch) == lane quad g.
// ---------------------------------------------------------------------------
__global__ void k_final(const float* __restrict__ x, const float* __restrict__ xw,
                        const float* __restrict__ agg, const float* __restrict__ as_,
                        const float* __restrict__ ad_, const float* __restrict__ mx,
                        const float* __restrict__ dn, const float* __restrict__ bias,
                        const float* __restrict__ g1w, const float* __restrict__ g1b,
                        const float* __restrict__ g2w, const float* __restrict__ g2b,
                        float* __restrict__ out, int Nn) {
  const int wid  = (int)((blockIdx.x * (size_t)blockDim.x + threadIdx.x) >> 5);
  const int lane = threadIdx.x & 31;
  if (wid >= Nn) return;

  const float es    = lrelu(as_[wid] + ad_[wid]);
  const float wself = expf(es - mx[wid]);
  const float inv   = 1.0f / (dn[wid] + wself + 1e-16f);

  const float4 va = *(const float4*)(agg + (size_t)wid * CCH + lane * 4);
  const float4 vw = *(const float4*)(xw  + (size_t)wid * CCH + lane * 4);
  const float4 vx = *(const float4*)(x   + (size_t)wid * CCH + lane * 4);
  const float4 vb = *(const float4*)(bias + lane * 4);

  float o[4];
  o[0] = (va.x + wself * vw.x) * inv + vb.x;
  o[1] = (va.y + wself * vw.y) * inv + vb.y;
  o[2] = (va.z + wself * vw.z) * inv + vb.z;
  o[3] = (va.w + wself * vw.w) * inv + vb.w;

  // GroupNorm 1 (quad reduction over 16 channels)
  float s1 = o[0] + o[1] + o[2] + o[3];
  float q1 = o[0]*o[0] + o[1]*o[1] + o[2]*o[2] + o[3]*o[3];
  s1 += __shfl_xor(s1, 1); s1 += __shfl_xor(s1, 2);
  q1 += __shfl_xor(q1, 1); q1 += __shfl_xor(q1, 2);
  float mu  = s1 * (1.0f / 16.0f);
  float var = q1 * (1.0f / 16.0f) - mu * mu;
  float rs  = rsqrtf(var + 1e-5f);
  const float4 w1 = *(const float4*)(g1w + lane * 4);
  const float4 b1 = *(const float4*)(g1b + lane * 4);
  float h[4];
  h[0] = (o[0] - mu) * rs * w1.x + b1.x;
  h[1] = (o[1] - mu) * rs * w1.y + b1.y;
  h[2] = (o[2] - mu) * rs * w1.z + b1.z;
  h[3] = (o[3] - mu) * rs * w1.w + b1.w;

  // exact-erf GELU + residual
  const float xres[4] = {vx.x, vx.y, vx.z, vx.w};
  float r[4];
#pragma unroll
  for (int i = 0; i < 4; ++i) {
    float g = 0.5f * h[i] * (1.0f + erff(h[i] * 0.70710678118654752f));
    r[i] = g + xres[i];
  }

  // GroupNorm 2
  float s2 = r[0] + r[1] + r[2] + r[3];
  float q2 = r[0]*r[0] + r[1]*r[1] + r[2]*r[2] + r[3]*r[3];
  s2 += __shfl_xor(s2, 1); s2 += __shfl_xor(s2, 2);
  q2 += __shfl_xor(q2, 1); q2 += __shfl_xor(q2, 2);
  float mu2  = s2 * (1.0f / 16.0f);
  float var2 = q2 * (1.0f / 16.0f) - mu2 * mu2;
  float rs2  = rsqrtf(var2 + 1e-5f);
  const float4 w2 = *(const float4*)(g2w + lane * 4);
  const float4 b2 = *(const float4*)(g2b + lane * 4);

  float4 res;
  res.x = (r[0] - mu2) * rs2 * w2.x + b2.x;
  res.y = (r[1] - mu2) * rs2 * w2.y + b2.y;
  res.z = (r[2] - mu2) * rs2 * w2.z + b2.z;
  res.w = (r[3] - mu2) * rs2 * w2.w + b2.w;
  *(float4*)(out + (size_t)wid * CCH + lane * 4) = res;
}

// ---------------------------------------------------------------------------
extern "C" void kernel_launch(void* const* d_in, const int* in_sizes, int n_in,
                              void* d_out, int out_size, void* d_ws, size_t ws_size,
                              hipStream_t stream) {
  const float*     x       = (const float*)d_in[0];
  const long long* ei      = (const long long*)d_in[1];
  const float*     W       = (const float*)d_in[2];
  const float*     att_src = (const float*)d_in[3];
  const float*     att_dst = (const float*)d_in[4];
  const float*     bias    = (const float*)d_in[5];
  const float*     g1w     = (const float*)d_in[6];
  const float*     g1b     = (const float*)d_in[7];
  const float*     g2w     = (const float*)d_in[8];
  const float*     g2b     = (const float*)d_in[9];
  float*           out     = (float*)d_out;

  const int Nn = in_sizes[0] / CCH;
  const int En = in_sizes[1] / 2;
  const long long* src = ei;
  const long long* dst = ei + En;

  float* ws  = (float*)d_ws;
  float* xw  = ws;
  float* agg = xw  + (size_t)Nn * CCH;
  float* as_ = agg + (size_t)Nn * CCH;
  float* ad_ = as_ + Nn;
  float* mx  = ad_ + Nn;
  float* dn  = mx  + Nn;

  k_gemm<<<(Nn + 127) / 128, 256, 0, stream>>>(x, W, xw, Nn);
  k_prep<<<(Nn + 7) / 8, 256, 0, stream>>>(xw, att_src, att_dst, as_, ad_, mx, dn, agg, Nn);
  k_emax<<<(En + 255) / 256, 256, 0, stream>>>(src, dst, as_, ad_, mx, En);
  k_escatter<<<(En + 7) / 8, 256, 0, stream>>>(src, dst, as_, ad_, mx, dn, xw, agg, En);
  k_final<<<(Nn + 7) / 8, 256, 0, stream>>>(x, xw, agg, as_, ad_, mx, dn, bias,
                                            g1w, g1b, g2w, g2b, out, Nn);
}